// WarpController_83674552860741
// MI455X (gfx1250) — compile-verified
//
#include <hip/hip_runtime.h>
#include <math.h>

// Problem constants (from reference setup_inputs)
#define B_    16
#define C_    64
#define H_    112
#define W_    112
#define F_    (C_*H_*W_)        // 802816
#define NOUT  128
#define NL    16                // landmarks
#define HW    (H_*W_)           // 12544
#define IMG_OUT ((size_t)B_*C_*HW)  // 12845056 floats

#define SPLITK 224
#define KPER   (F_/SPLITK)      // 3584
#define KC     64               // K chunk staged in LDS
#define LSTR   68               // LDS row stride (floats): KC + 4 pad -> conflict-free ds_load_b64

typedef __attribute__((ext_vector_type(2)))  float v2f;
typedef __attribute__((ext_vector_type(8)))  float v8f;
typedef int v4i_ __attribute__((vector_size(16)));   // matches builtin's int4 param

// workspace layout (float offsets)
#define WS_OUTFEAT 0            // 2048 floats: out = feat@W1.T + b1
#define WS_DST     2048         // 512: dst control points (B,16,2)
#define WS_DISP    2560         // 512: flows_ctrl = dst - src
#define WS_SOL     3072         // 608: TPS solution per batch (16x2 w, 3x2 v)
#define WS_PART    4096         // SPLITK*2048 GEMM partials

// ---------------------------------------------------------------------------
// Async global->LDS staging (CDNA5 GLOBAL_LOAD_ASYNC_TO_LDS_B128, ASYNCcnt).
// Signature per hipcc diagnostic: param0 = AS1 int4*, then AS3 int4*, imm, imm.
// ---------------------------------------------------------------------------
#if __has_builtin(__builtin_amdgcn_global_load_async_to_lds_b128)
__device__ __forceinline__ void async_copy16(const float* g, float* l) {
    __builtin_amdgcn_global_load_async_to_lds_b128(
        (__attribute__((address_space(1))) v4i_*)(g),
        (__attribute__((address_space(3))) v4i_*)(l),
        0, 0);
}
__device__ __forceinline__ void async_wait0() {
#if __has_builtin(__builtin_amdgcn_s_wait_asynccnt)
    __builtin_amdgcn_s_wait_asynccnt(0);
#else
    asm volatile("s_wait_asynccnt 0" ::: "memory");
#endif
}
#else
__device__ __forceinline__ void async_copy16(const float* g, float* l) {
    *(float4*)l = *(const float4*)g;
}
__device__ __forceinline__ void async_wait0() {}
#endif

// ---------------------------------------------------------------------------
// Kernel 1: split-K WMMA GEMM with double-buffered LDS staging.
// Block = 256 threads (8 waves); wave w computes N-tile [w*16, w*16+16).
// out[m,n] = sum_k feat[m,k] * W1[n,k]
// ---------------------------------------------------------------------------
__global__ __launch_bounds__(256)
void gemm1_wmma(const float* __restrict__ x, const float* __restrict__ W1,
                float* __restrict__ part) {
    __shared__ float sA[2][16  * LSTR];   // x tile:  16 x KC
    __shared__ float sB[2][128 * LSTR];   // W1 tile: 128 x KC

    const int t     = threadIdx.x;
    const int lane  = t & 31;
    const int wave  = t >> 5;             // 0..7 -> n0 = wave*16
    const int split = blockIdx.x;
    const int kbeg  = split * KPER;
    const int half  = lane >> 4;          // 0: lanes 0-15, 1: lanes 16-31
    const int l15   = lane & 15;
    const int koff  = half * 2;           // f32 frag K offset per lane half

    // staging map: 16 consecutive lanes cover 256 contiguous bytes of one row
    const int srow = t >> 4;              // 0..15
    const int scol = (t & 15) * 4;        // 0,4,...,60

    const float* xrow = x  + (size_t)srow * F_ + kbeg;
    const float* wrow = W1 + (size_t)srow * F_ + kbeg;   // rows srow + 16p

    v8f acc = {};

    // prologue: stage chunk 0 into buffer 0
    async_copy16(xrow + scol, &sA[0][srow * LSTR + scol]);
#pragma unroll
    for (int p = 0; p < 8; ++p)
        async_copy16(wrow + (size_t)(16 * p) * F_ + scol,
                     &sB[0][(16 * p + srow) * LSTR + scol]);
    async_wait0();
    __syncthreads();

    int cur = 0;
    for (int kc = 0; kc < KPER; kc += KC) {
        const int nxt = cur ^ 1;
        // prefetch next chunk into the other buffer (async, overlaps WMMA)
        if (kc + KC < KPER) {
            async_copy16(xrow + kc + KC + scol, &sA[nxt][srow * LSTR + scol]);
#pragma unroll
            for (int p = 0; p < 8; ++p)
                async_copy16(wrow + (size_t)(16 * p) * F_ + kc + KC + scol,
                             &sB[nxt][(16 * p + srow) * LSTR + scol]);
        }
        // compute on current buffer: 16 WMMAs per chunk
        const float* la = &sA[cur][l15 * LSTR + koff];
        const float* lb = &sB[cur][(wave * 16 + l15) * LSTR + koff];
#pragma unroll
        for (int k = 0; k < KC; k += 4) {
            float2 av = *(const float2*)(la + k);
            float2 bv = *(const float2*)(lb + k);
            v2f a; a.x = av.x; a.y = av.y;
            v2f b; b.x = bv.x; b.y = bv.y;
            acc = __builtin_amdgcn_wmma_f32_16x16x4_f32(
                false, a, false, b, (short)0, acc, false, false);
        }
        async_wait0();
        __syncthreads();
        cur = nxt;
    }

    // C/D layout: VGPR r -> M = r + half*8, N = wave*16 + l15
    float* o = part + (size_t)split * 2048 + (size_t)(half * 8) * NOUT
             + wave * 16 + l15;
#pragma unroll
    for (int r = 0; r < 8; ++r) o[(size_t)r * NOUT] = acc[r];
}

// ---------------------------------------------------------------------------
// Kernel 2: deterministic split-K reduction + bias.
// ---------------------------------------------------------------------------
__global__ __launch_bounds__(256)
void reduce_bias(const float* __restrict__ part, const float* __restrict__ b1,
                 float* __restrict__ outfeat) {
    int idx = blockIdx.x * blockDim.x + threadIdx.x;   // 0..2047
    if (idx >= B_ * NOUT) return;
    float s = 0.f;
    for (int k = 0; k < SPLITK; ++k) s += part[(size_t)k * 2048 + idx];
    outfeat[idx] = s + b1[idx & (NOUT - 1)];
}

// ---------------------------------------------------------------------------
// Kernel 3: landmark head.  pred = out@W2.T+b2+mean; disp = (out@W3.T+b3)*scale
// ---------------------------------------------------------------------------
__global__ __launch_bounds__(512)
void head(const float* __restrict__ outfeat,
          const float* __restrict__ W2, const float* __restrict__ b2,
          const float* __restrict__ W3, const float* __restrict__ b3,
          const float* __restrict__ lm_mean, const float* __restrict__ scales,
          float* __restrict__ d_out, float* __restrict__ ws_dst,
          float* __restrict__ ws_disp) {
    __shared__ float sOut[B_ * NOUT];
    __shared__ float sSq[512];
    __shared__ float sNorm[B_];

    int t = threadIdx.x;
    for (int i = t; i < B_ * NOUT; i += 512) sOut[i] = outfeat[i];
    __syncthreads();

    int b = t >> 5, j = t & 31;
    const float* o  = sOut + b * NOUT;
    const float* w2 = W2 + j * NOUT;
    const float* w3 = W3 + j * NOUT;
    float d2 = 0.f, d3 = 0.f;
    for (int i = 0; i < NOUT; ++i) { d2 += o[i] * w2[i]; d3 += o[i] * w3[i]; }

    float pred = d2 + b2[j] + lm_mean[j];
    float disp = (d3 + b3[j]) * scales[b];

    d_out[IMG_OUT + b * 32 + j] = pred;         // landmarks_pred
    ws_dst[b * 32 + j]  = pred + disp;          // dst control points
    ws_disp[b * 32 + j] = disp;                 // flows_ctrl

    sSq[t] = disp * disp;
    __syncthreads();
    if (j == 0) {
        float s = 0.f;
        for (int q = 0; q < 32; ++q) s += sSq[b * 32 + q];
        sNorm[b] = sqrtf(s);
    }
    __syncthreads();
    if (t == 0) {
        float s = 0.f;
        for (int q = 0; q < B_; ++q) s += sNorm[q];
        d_out[IMG_OUT + 512] = s / (float)B_;   // landmarks_norm
    }
}

// ---------------------------------------------------------------------------
// Kernel 4: per-batch TPS solve (19x19, partial pivoting Gauss-Jordan).
// ---------------------------------------------------------------------------
#define NP 19
__global__ __launch_bounds__(32)
void tps_solve(const float* __restrict__ ws_dst, const float* __restrict__ ws_disp,
               float* __restrict__ ws_sol) {
    __shared__ float M[NP][NP + 2];
    int b = blockIdx.x, t = threadIdx.x;
    const float* pts = ws_dst  + b * 32;
    const float* val = ws_disp + b * 32;

    for (int idx = t; idx < NP * (NP + 2); idx += 32) {
        int i = idx / (NP + 2), j = idx % (NP + 2);
        float vv;
        if (j < NP) {
            if (i < NL && j < NL) {
                float dy = pts[2 * i]     - pts[2 * j];
                float dx = pts[2 * i + 1] - pts[2 * j + 1];
                float r2 = dy * dy + dx * dx;
                vv = 0.5f * r2 * logf(fmaxf(r2, 1e-10f));
                if (i == j) vv += 1e-6f;
            } else if (i < NL) {
                vv = (j == 16) ? pts[2 * i] : (j == 17) ? pts[2 * i + 1] : 1.f;
            } else if (j < NL) {
                vv = (i == 16) ? pts[2 * j] : (i == 17) ? pts[2 * j + 1] : 1.f;
            } else vv = 0.f;
        } else {
            int d = j - NP;
            vv = (i < NL) ? val[2 * i + d] : 0.f;
        }
        M[i][j] = vv;
    }
    __syncthreads();

    for (int k = 0; k < NP; ++k) {
        if (t == 0) {
            int piv = k; float mx = fabsf(M[k][k]);
            for (int r = k + 1; r < NP; ++r) {
                float a = fabsf(M[r][k]);
                if (a > mx) { mx = a; piv = r; }
            }
            if (piv != k)
                for (int j = 0; j < NP + 2; ++j) {
                    float tmp = M[k][j]; M[k][j] = M[piv][j]; M[piv][j] = tmp;
                }
            float inv = 1.0f / M[k][k];
            for (int j = 0; j < NP + 2; ++j) M[k][j] *= inv;
        }
        __syncthreads();
        for (int i = t; i < NP; i += 32) {
            if (i != k) {
                float f = M[i][k];
                for (int j = k; j < NP + 2; ++j) M[i][j] -= f * M[k][j];
            }
        }
        __syncthreads();
    }
    for (int idx = t; idx < NP * 2; idx += 32)
        ws_sol[b * 38 + idx] = M[idx >> 1][NP + (idx & 1)];
}

// ---------------------------------------------------------------------------
// Kernel 5: dense flow eval + bilinear warp + NHWC->NCHW transpose.
// ---------------------------------------------------------------------------
__global__ __launch_bounds__(256)
void warp_sample(const float* __restrict__ img, const float* __restrict__ ws_dst,
                 const float* __restrict__ ws_sol, float* __restrict__ out) {
    __shared__ float sty[NL], stx[NL], swy[NL], swx[NL], sv[6];
    int b = blockIdx.x / 49;
    int pix = (blockIdx.x % 49) * 256 + threadIdx.x;

    if (threadIdx.x < NL) {
        sty[threadIdx.x] = ws_dst[b * 32 + 2 * threadIdx.x];
        stx[threadIdx.x] = ws_dst[b * 32 + 2 * threadIdx.x + 1];
        swy[threadIdx.x] = ws_sol[b * 38 + 2 * threadIdx.x];
        swx[threadIdx.x] = ws_sol[b * 38 + 2 * threadIdx.x + 1];
    }
    if (threadIdx.x < 6) sv[threadIdx.x] = ws_sol[b * 38 + 32 + threadIdx.x];
    __syncthreads();

    int gy = pix / W_, gx = pix % W_;
    float fy = (float)gy, fx = (float)gx;
    float fl_y = fy * sv[0] + fx * sv[2] + sv[4];
    float fl_x = fy * sv[1] + fx * sv[3] + sv[5];
#pragma unroll
    for (int i = 0; i < NL; ++i) {
        float dy = fy - sty[i], dx = fx - stx[i];
        float r2 = dy * dy + dx * dx;
        float ph = 0.5f * r2 * logf(fmaxf(r2, 1e-10f));
        fl_y += ph * swy[i];
        fl_x += ph * swx[i];
    }
    float qy = fy - fl_y, qx = fx - fl_x;

    int y0 = (int)fminf(fmaxf(floorf(qy), 0.f), (float)(H_ - 2));
    int x0 = (int)fminf(fmaxf(floorf(qx), 0.f), (float)(W_ - 2));
    float ay = fminf(fmaxf(qy - (float)y0, 0.f), 1.f);
    float ax = fminf(fmaxf(qx - (float)x0, 0.f), 1.f);
    float w00 = (1.f - ay) * (1.f - ax), w01 = (1.f - ay) * ax;
    float w10 = ay * (1.f - ax),         w11 = ay * ax;

    const float* p00 = img + (((size_t)b * H_ + y0) * W_ + x0) * C_;
    const float* p01 = p00 + C_;
    const float* p10 = p00 + (size_t)W_ * C_;
    const float* p11 = p10 + C_;
    float* ob = out + (size_t)b * C_ * HW + pix;

#pragma unroll 4
    for (int c = 0; c < C_; c += 4) {
        float4 a0 = *(const float4*)(p00 + c);
        float4 a1 = *(const float4*)(p01 + c);
        float4 a2 = *(const float4*)(p10 + c);
        float4 a3 = *(const float4*)(p11 + c);
        ob[(size_t)(c + 0) * HW] = w00 * a0.x + w01 * a1.x + w10 * a2.x + w11 * a3.x;
        ob[(size_t)(c + 1) * HW] = w00 * a0.y + w01 * a1.y + w10 * a2.y + w11 * a3.y;
        ob[(size_t)(c + 2) * HW] = w00 * a0.z + w01 * a1.z + w10 * a2.z + w11 * a3.z;
        ob[(size_t)(c + 3) * HW] = w00 * a0.w + w01 * a1.w + w10 * a2.w + w11 * a3.w;
    }
}

// ---------------------------------------------------------------------------
extern "C" void kernel_launch(void* const* d_in, const int* in_sizes, int n_in,
                              void* d_out, int out_size, void* d_ws, size_t ws_size,
                              hipStream_t stream) {
    const float* x      = (const float*)d_in[0];
    const float* img    = (const float*)d_in[1];
    const float* scales = (const float*)d_in[2];
    const float* W1     = (const float*)d_in[3];
    const float* b1     = (const float*)d_in[4];
    const float* W2     = (const float*)d_in[5];
    const float* b2     = (const float*)d_in[6];
    const float* W3     = (const float*)d_in[7];
    const float* b3     = (const float*)d_in[8];
    const float* lm     = (const float*)d_in[9];
    float* out = (float*)d_out;
    float* ws  = (float*)d_ws;

    float* outfeat = ws + WS_OUTFEAT;
    float* wdst    = ws + WS_DST;
    float* wdisp   = ws + WS_DISP;
    float* wsol    = ws + WS_SOL;
    float* part    = ws + WS_PART;

    gemm1_wmma<<<SPLITK, 256, 0, stream>>>(x, W1, part);
    reduce_bias<<<8, 256, 0, stream>>>(part, b1, outfeat);
    head<<<1, 512, 0, stream>>>(outfeat, W2, b2, W3, b3, lm, scales,
                                out, wdst, wdisp);
    tps_solve<<<B_, 32, 0, stream>>>(wdst, wdisp, wsol);
    warp_sample<<<B_ * 49, 256, 0, stream>>>(img, wdst, wsol, out);

    (void)in_sizes; (void)n_in; (void)out_size; (void)ws_size;
}